// TFnet_36172214567113
// MI455X (gfx1250) — compile-verified
//
#include <hip/hip_runtime.h>
#include <hip/hip_bf16.h>
#include <math.h>

// ---------------------------------------------------------------------------
// TFnet forward on gfx1250 (CDNA5, wave32).
// Core trick: sh[b, q*16+m] = A[b,q] * C[b,m]  (rank-structured), so the big
// (4096 x 11856) @ (11856 x 400) GEMM generates its A-operand on the fly from
// a 12 MB A matrix + 256 KB C matrix instead of a 194 MB sh matrix.
// All GEMMs use V_WMMA_F32_16X16X4_F32 (exact fp32, matches reference).
// ---------------------------------------------------------------------------

typedef float v2f __attribute__((ext_vector_type(2)));
typedef float v8f __attribute__((ext_vector_type(8)));

#define NB    4096      // batch
#define NQ    741       // triu pairs of 39 fields
#define NF    39
#define NFD   624       // 39*16
#define NK    400       // DEEP
#define NQM   11856     // NQ*16
#define NV    100000
#define QPAD  745       // 745 mod 64 = 41, coprime with 64 -> conflict-free rows
#define FPAD  625

__device__ __forceinline__ v8f wmma4(v2f a, v2f b, v8f c) {
    // D = A(16x4 f32) * B(4x16 f32) + C(16x16 f32)
    return __builtin_amdgcn_wmma_f32_16x16x4_f32(
        false, a, false, b, (short)0, c, false, false);
}

// ---------------------------------------------------------------------------
// K0: tiny prep — T3diag (16x16), rowsum(T2) (16), triu pair index tables.
// ---------------------------------------------------------------------------
__global__ __launch_bounds__(256) void k_prep(const float* __restrict__ T3,
                                              const float* __restrict__ T2,
                                              float* __restrict__ t3d,
                                              float* __restrict__ t2rs,
                                              int* __restrict__ leftI,
                                              int* __restrict__ rightI) {
    int tid = threadIdx.x;
    if (tid < 256) {                         // T3diag[d,m] = T3[d, m, d]
        int d = tid >> 4, m = tid & 15;
        t3d[d * 16 + m] = T3[d * 256 + m * 16 + d];
    }
    if (tid < 16) {                          // rowsum over 256
        float s = 0.f;
        for (int k = 0; k < 256; ++k) s += T2[tid * 256 + k];
        t2rs[tid] = s;
    }
    if (tid == 0) {                          // np.triu_indices(39, k=1) order
        int q = 0;
        for (int i = 0; i < NF; ++i)
            for (int j = i + 1; j < NF; ++j) { leftI[q] = i; rightI[q] = j; ++q; }
    }
}

// ---------------------------------------------------------------------------
// K1: build fm2 (B, 39, 16): numeric fields = xv * emb[f,0,:], cat = gather.
// ---------------------------------------------------------------------------
__global__ __launch_bounds__(640) void k_fm2(const int* __restrict__ xi,
                                             const float* __restrict__ xv,
                                             const float* __restrict__ emb,
                                             float* __restrict__ fm2) {
    int b = blockIdx.x, t = threadIdx.x;
    if (t >= NFD) return;
    int f = t >> 4, d = t & 15;
    float v;
    if (f < 13) {
        v = xv[b * 13 + f] * emb[(size_t)f * NV * 16 + d];
    } else {
        int idx = xi[b * 26 + (f - 13)];
        v = emb[((size_t)f * NV + (size_t)idx) * 16 + d];
    }
    fm2[(size_t)b * NFD + t] = v;
}

// ---------------------------------------------------------------------------
// K2: per-sample: A[b,q] = (vi.vj)*gc[q]  (full Gram over pairs) and
//     C[b,m] = softmax_m( pairsum . T3diag ) * rowsum(T2)[m]
//     pairsum via FM identity: sum_{i<j} x_i x_j = 0.5*((sum x)^2 - sum x^2)
// ---------------------------------------------------------------------------
__global__ __launch_bounds__(256) void k_stats(const float* __restrict__ fm2,
                                               const float* __restrict__ gc,
                                               const float* __restrict__ t3d,
                                               const float* __restrict__ t2rs,
                                               const int* __restrict__ leftI,
                                               const int* __restrict__ rightI,
                                               float* __restrict__ A,
                                               float* __restrict__ Ct) {
    __shared__ float Fs[NFD];
    __shared__ float ps[16];
    __shared__ float sga[16];
    int b = blockIdx.x, tid = threadIdx.x;
    const float* src = fm2 + (size_t)b * NFD;
    for (int i = tid; i < NFD; i += 256) Fs[i] = src[i];
    __syncthreads();

    // all pairwise dots, scaled by gc
    for (int q = tid; q < NQ; q += 256) {
        int i = leftI[q], j = rightI[q];
        float dot = 0.f;
        #pragma unroll
        for (int d = 0; d < 16; ++d) dot += Fs[i * 16 + d] * Fs[j * 16 + d];
        A[(size_t)b * NQ + q] = dot * gc[q];
    }

    if (tid < 16) {                          // FM identity per d-slot
        float S = 0.f, SS = 0.f;
        #pragma unroll
        for (int f = 0; f < NF; ++f) { float v = Fs[f * 16 + tid]; S += v; SS += v * v; }
        ps[tid] = 0.5f * (S * S - SS);
    }
    __syncthreads();
    if (tid < 16) {                          // ga[m] = ps . T3diag[:,m]
        float g = 0.f;
        #pragma unroll
        for (int d = 0; d < 16; ++d) g += ps[d] * t3d[d * 16 + tid];
        sga[tid] = g;
    }
    __syncthreads();
    if (tid < 16) {                          // softmax over 16, scale by t2rs
        float mx = sga[0];
        #pragma unroll
        for (int i = 1; i < 16; ++i) mx = fmaxf(mx, sga[i]);
        float denom = 0.f;
        #pragma unroll
        for (int i = 0; i < 16; ++i) denom += expf(sga[i] - mx);
        Ct[b * 16 + tid] = (expf(sga[tid] - mx) / denom) * t2rs[tid];
    }
}

// ---------------------------------------------------------------------------
// K3: big fused GEMM: th_pre[b,k] = sum_{q,m} A[b,q]*C[b,m]*fi_w[q*16+m, k] + fi_b[k]
// Block = 16 batch rows, 25 waves; wave w owns k-columns [16w, 16w+16).
// A-fragment generated on the fly: sh[b, j] = As[b][j>>4] * Cs[b][j&15].
// f32 WMMA operand layout (ISA 7.12.2): lane L -> row/col L%16;
//   lanes 0-15 carry K={j,j+1}, lanes 16-31 carry K={j+2,j+3}.
// ---------------------------------------------------------------------------
__global__ __launch_bounds__(800) void k_figemm(const float* __restrict__ A,
                                                const float* __restrict__ Ct,
                                                const float* __restrict__ W,
                                                const float* __restrict__ bias,
                                                float* __restrict__ outp) {
    __shared__ float As[16][QPAD];
    __shared__ float Cs[16][17];
    int tb = blockIdx.x, tid = threadIdx.x;
    for (int i = tid; i < 16 * NQ; i += 800) {
        int r = i / NQ, q = i - r * NQ;
        As[r][q] = A[(size_t)(tb * 16 + r) * NQ + q];
    }
    if (tid < 256)
        Cs[tid >> 4][tid & 15] = Ct[(tb * 16 + (tid >> 4)) * 16 + (tid & 15)];
    __syncthreads();

    int wave = tid >> 5, lane = tid & 31;
    int row  = lane & 15;
    int koff = (lane >> 4) << 1;             // 0 for lanes 0-15, 2 for 16-31
    int kcol = wave * 16 + row;              // N index (== output column)
    v8f acc = {};
    #pragma unroll 4
    for (int j = 0; j < NQM; j += 4) {
        int j0 = j + koff, j1 = j0 + 1;
        v2f a, bf;
        a.x  = As[row][j0 >> 4] * Cs[row][j0 & 15];   // sh on the fly
        a.y  = As[row][j1 >> 4] * Cs[row][j1 & 15];
        bf.x = W[(size_t)j0 * NK + kcol];
        bf.y = W[(size_t)j1 * NK + kcol];
        acc  = wmma4(a, bf, acc);
    }
    float bs = bias[kcol];
    int mbase = (lane >> 4) << 3;            // D layout: VGPR r -> M = r (+8 hi)
    #pragma unroll
    for (int r = 0; r < 8; ++r)
        outp[(size_t)(tb * 16 + mbase + r) * NK + kcol] = acc[r] + bs;
}

// ---------------------------------------------------------------------------
// K4: emb GEMM: emb_pre = fm2.reshape(B, 624) @ emb_w + emb_b, same structure.
// ---------------------------------------------------------------------------
__global__ __launch_bounds__(800) void k_embgemm(const float* __restrict__ fm2,
                                                 const float* __restrict__ W,
                                                 const float* __restrict__ bias,
                                                 float* __restrict__ outp) {
    __shared__ float Fs[16][FPAD];
    int tb = blockIdx.x, tid = threadIdx.x;
    for (int i = tid; i < 16 * NFD; i += 800) {
        int r = i / NFD, c = i - r * NFD;
        Fs[r][c] = fm2[(size_t)(tb * 16 + r) * NFD + c];
    }
    __syncthreads();

    int wave = tid >> 5, lane = tid & 31;
    int row  = lane & 15;
    int koff = (lane >> 4) << 1;
    int kcol = wave * 16 + row;
    v8f acc = {};
    #pragma unroll 4
    for (int j = 0; j < NFD; j += 4) {
        int j0 = j + koff;
        v2f a, bf;
        a.x  = Fs[row][j0];
        a.y  = Fs[row][j0 + 1];
        bf.x = W[(size_t)j0 * NK + kcol];
        bf.y = W[(size_t)(j0 + 1) * NK + kcol];
        acc  = wmma4(a, bf, acc);
    }
    float bs = bias[kcol];
    int mbase = (lane >> 4) << 3;
    #pragma unroll
    for (int r = 0; r < 8; ++r)
        outp[(size_t)(tb * 16 + mbase + r) * NK + kcol] = acc[r] + bs;
}

// ---------------------------------------------------------------------------
// K5: BN column stats folded into per-column affine: y = a*x + b2.
// grid (400, 2): y==0 -> fi branch, y==1 -> emb branch.
// coefs: [0..399]=fiA, [400..799]=fiB, [800..1199]=emA, [1200..1599]=emB.
// ---------------------------------------------------------------------------
__global__ __launch_bounds__(256) void k_bn(const float* __restrict__ thp,
                                            const float* __restrict__ emp,
                                            const float* __restrict__ fig,
                                            const float* __restrict__ fibt,
                                            const float* __restrict__ emg,
                                            const float* __restrict__ embt,
                                            float* __restrict__ coefs) {
    int col = blockIdx.x, sel = blockIdx.y, tid = threadIdx.x;
    const float* src = sel ? emp : thp;
    __shared__ float rs[256], rss[256];
    float s = 0.f, ss = 0.f;
    for (int r = tid; r < NB; r += 256) {
        float v = src[(size_t)r * NK + col];
        s += v; ss += v * v;
    }
    rs[tid] = s; rss[tid] = ss;
    __syncthreads();
    for (int off = 128; off > 0; off >>= 1) {
        if (tid < off) { rs[tid] += rs[tid + off]; rss[tid] += rss[tid + off]; }
        __syncthreads();
    }
    if (tid == 0) {
        float mean = rs[0] * (1.f / NB);
        float var  = rss[0] * (1.f / NB) - mean * mean;
        float g  = sel ? emg[col] : fig[col];
        float bt = sel ? embt[col] : fibt[col];
        float a  = g * rsqrtf(var + 1e-5f);
        coefs[sel * 800 + col]       = a;
        coefs[sel * 800 + 400 + col] = bt - mean * a;
    }
}

// ---------------------------------------------------------------------------
// K6: out[b] = sigmoid( relu(aff(th)) . fc_w[0:400] + relu(aff(emb)) . fc_w[400:800] + fc_b )
// ---------------------------------------------------------------------------
__global__ __launch_bounds__(256) void k_final(const float* __restrict__ thp,
                                               const float* __restrict__ emp,
                                               const float* __restrict__ coefs,
                                               const float* __restrict__ fcw,
                                               const float* __restrict__ fcb,
                                               float* __restrict__ out) {
    int b = blockIdx.x, tid = threadIdx.x;
    __shared__ float rs[256];
    float acc = 0.f;
    for (int k = tid; k < NK; k += 256) {
        float t = fmaxf(coefs[k] * thp[(size_t)b * NK + k] + coefs[400 + k], 0.f);
        acc += t * fcw[k];
        float e = fmaxf(coefs[800 + k] * emp[(size_t)b * NK + k] + coefs[1200 + k], 0.f);
        acc += e * fcw[400 + k];
    }
    rs[tid] = acc;
    __syncthreads();
    for (int off = 128; off > 0; off >>= 1) {
        if (tid < off) rs[tid] += rs[tid + off];
        __syncthreads();
    }
    if (tid == 0) {
        float x = rs[0] + fcb[0];
        out[b] = 1.f / (1.f + expf(-x));
    }
}

// ---------------------------------------------------------------------------
// Workspace layout (floats):
//   0        T3diag(256) | 256 T2rowsum(16) | 288 coefs(1600) | 1888 pairs(1482 ints)
//   4096     fm2  (4096*624)
//   2560000  A    (4096*741)
//   5595136  Ct   (4096*16)
//   5660672  th_pre (4096*400)
//   7299072  emb_pre(4096*400)   -> total 8937472 floats (~35.8 MB)
// ---------------------------------------------------------------------------
extern "C" void kernel_launch(void* const* d_in, const int* in_sizes, int n_in,
                              void* d_out, int out_size, void* d_ws, size_t ws_size,
                              hipStream_t stream) {
    (void)in_sizes; (void)n_in; (void)out_size; (void)ws_size;
    const int*   xi        = (const int*)  d_in[0];
    const float* xv        = (const float*)d_in[1];
    const float* emb_t     = (const float*)d_in[2];
    const float* T3        = (const float*)d_in[3];
    const float* T2        = (const float*)d_in[4];
    const float* gc        = (const float*)d_in[5];
    const float* fi_w      = (const float*)d_in[6];
    const float* fi_b      = (const float*)d_in[7];
    const float* fi_gamma  = (const float*)d_in[8];
    const float* fi_beta   = (const float*)d_in[9];
    const float* emb_w     = (const float*)d_in[10];
    const float* emb_b     = (const float*)d_in[11];
    const float* emb_gamma = (const float*)d_in[12];
    const float* emb_beta  = (const float*)d_in[13];
    const float* fc_w      = (const float*)d_in[14];
    const float* fc_b      = (const float*)d_in[15];
    float* out = (float*)d_out;
    float* ws  = (float*)d_ws;

    float* t3d   = ws;
    float* t2rs  = ws + 256;
    float* coefs = ws + 288;
    int*   leftI = (int*)(ws + 1888);
    int*   rightI= leftI + NQ;
    float* fm2   = ws + 4096;
    float* A     = ws + 2560000;
    float* Ct    = ws + 5595136;
    float* thp   = ws + 5660672;
    float* emp   = ws + 7299072;

    k_prep<<<1, 256, 0, stream>>>(T3, T2, t3d, t2rs, leftI, rightI);
    k_fm2<<<NB, 640, 0, stream>>>(xi, xv, emb_t, fm2);
    k_stats<<<NB, 256, 0, stream>>>(fm2, gc, t3d, t2rs, leftI, rightI, A, Ct);
    k_figemm<<<NB / 16, 800, 0, stream>>>(A, Ct, fi_w, fi_b, thp);
    k_embgemm<<<NB / 16, 800, 0, stream>>>(fm2, emb_w, emb_b, emp);
    k_bn<<<dim3(NK, 2), 256, 0, stream>>>(thp, emp, fi_gamma, fi_beta,
                                          emb_gamma, emb_beta, coefs);
    k_final<<<NB, 256, 0, stream>>>(thp, emp, coefs, fc_w, fc_b, out);
}